// Instance3DGlobalScalingModel_87282325390109
// MI455X (gfx1250) — compile-verified
//
#include <hip/hip_runtime.h>
#include <math.h>

typedef __attribute__((ext_vector_type(2))) float v2f;
typedef __attribute__((ext_vector_type(8))) float v8f;

// One wave32 workgroup. Each 4x4 matmul in the chain is one
// V_WMMA_F32_16X16X4_F32 (K=4 exact; f32 accumulate == jnp.float32 semantics).
// All WMMA fragment accesses are unconditional: operands live in LDS buffers
// shaped like the wave32 fragment layouts, zero-padded outside the 4x4 tile.
// Setup is cooperative (all 32 lanes) wherever the pattern is lane-regular.
__global__ __launch_bounds__(32)
void Instance3DGlobalScalingModel_kernel(const float* __restrict__ header,
                                         const float* __restrict__ vol_affine,
                                         const float* __restrict__ cr,
                                         const float* __restrict__ normvec,
                                         const float* __restrict__ translation,
                                         const float* __restrict__ angle,
                                         const float* __restrict__ scaling,
                                         float* __restrict__ out) {
  // sB[m][k*16 + n] = M_m[k][n] for n<4, 0 for n>=4  (B-fragment shape)
  // sPD[r*16 + c]   = P[r][c]  for r,c<4, 0 elsewhere (A/D tile shape)
  __shared__ float sB[12][64];
  __shared__ float sPD[256];
  __shared__ float sVA[64];

  const int lane = (int)threadIdx.x;
  const int row  = lane & 15;
  const int hi   = lane >> 4;
  const int k0   = hi ? 2 : 0;
  const int k1   = k0 + 1;

  // ---- phase 1: cooperative fills (all lanes, unconditional) ----
  // zero the 16x16 product tile (8 floats per lane)
#pragma unroll
  for (int j = 0; j < 8; ++j) sPD[lane * 8 + j] = 0.0f;

  // vol_affine as a zero-padded B-fragment (2 entries per lane)
#pragma unroll
  for (int t = 0; t < 2; ++t) {
    int idx = lane + t * 32;
    int k = idx >> 4, n = idx & 15;
    float v = vol_affine[k * 4 + (n & 3)];   // unconditional, clamped address
    sVA[idx] = (n < 4) ? v : 0.0f;
  }

  // identity pattern is identical for every matrix: 2 entries per lane per matrix
  {
    const int ia = lane, ib = lane + 32;
    const float ida = ((ia >> 4) == (ia & 15)) ? 1.0f : 0.0f;
    const float idb = ((ib >> 4) == (ib & 15)) ? 1.0f : 0.0f;
#pragma unroll
    for (int m = 1; m <= 11; ++m) {
      sB[m][ia] = ida;
      sB[m][ib] = idb;
    }
  }
  __syncthreads();

  // ---- phase 2: sparse patches over the identity fill ----
  // header 4x4 corner, one element per lane (lanes 0..15)
  if (lane < 16) sB[11][(lane >> 2) * 16 + (lane & 3)] = header[lane];

  if (lane == 0) {
    // uniform scalars
    float n0 = normvec[0], n1 = normvec[1], n2 = normvec[2];
    float nn = sqrtf(n0 * n0 + n1 * n1 + n2 * n2);
    float v0 = n0 / nn, v1 = n1 / nn, v2 = n2 / nn;
    float nxy = sqrtf(v0 * v0 + v1 * v1);
    float a, b;
    if (nxy == 0.0f) { a = 1.0f; b = 0.0f; }
    else             { a = v0 / nxy; b = v1 / nxy; }
    float sxy = expf(scaling[0] * 0.01f);
    float sz  = expf(scaling[1] * 0.01f);
    const float d2r = 3.14159265358979323846f / 180.0f;
    float a0 = angle[0] * d2r, a1 = angle[1] * d2r, a2 = angle[2] * d2r;
    float c0 = cosf(a0), s0 = sinf(a0);
    float c1 = cosf(a1), s1 = sinf(a1);
    float c2 = cosf(a2), s2 = sinf(a2);

    // patches: element (r,c) lives at sB[m][r*16 + c]
    // 1: T1inv (+cr)
    sB[1][3] = cr[0]; sB[1][19] = cr[1]; sB[1][35] = cr[2];
    // 2: T2inv
    sB[2][0] = a;  sB[2][1] = -b; sB[2][16] = b;  sB[2][17] = a;
    // 3: T3inv
    sB[3][0] = v2; sB[3][2] = nxy; sB[3][32] = -nxy; sB[3][34] = v2;
    // 4: T4b3 (rot z)
    sB[4][0] = c2; sB[4][1] = -s2; sB[4][16] = s2; sB[4][17] = c2;
    // 5: T4b2 (rot y)
    sB[5][0] = c1; sB[5][2] = s1; sB[5][32] = -s1; sB[5][34] = c1;
    // 6: T4b1 (rot x)
    sB[6][17] = c0; sB[6][18] = -s0; sB[6][33] = s0; sB[6][34] = c0;
    // 7: T4a (scaling)
    sB[7][0] = sxy; sB[7][17] = sxy; sB[7][34] = sz;
    // 8: T3
    sB[8][0] = v2; sB[8][2] = -nxy; sB[8][32] = nxy; sB[8][34] = v2;
    // 9: T2
    sB[9][0] = a;  sB[9][1] = b; sB[9][16] = -b; sB[9][17] = a;
    // 10: T1 (-cr)
    sB[10][3] = -cr[0]; sB[10][19] = -cr[1]; sB[10][35] = -cr[2];

    // P initialized to Ttrans (identity + translation) in the zeroed tile
    sPD[0] = 1.0f; sPD[17] = 1.0f; sPD[34] = 1.0f; sPD[51] = 1.0f;
    sPD[3] = translation[0]; sPD[19] = translation[1]; sPD[35] = translation[2];
  }
  __syncthreads();

  // ---- chain: P = (((Ttrans @ T1inv) @ ...) @ T1) @ header, 11 WMMAs ----
#pragma unroll
  for (int i = 1; i <= 11; ++i) {
    v2f A, B;
    A.x = sPD[row * 16 + k0];
    A.y = sPD[row * 16 + k1];
    B.x = sB[i][k0 * 16 + row];
    B.y = sB[i][k1 * 16 + row];
    v8f C = {};
    v8f D = __builtin_amdgcn_wmma_f32_16x16x4_f32(false, A, false, B,
                                                  (short)0, C, false, false);
    __syncthreads();
    // Unconditional write-back: lanes<16 write rows 0..3 (data),
    // lanes>=16 write rows 8..11 (zeros) -- invariant preserved, no branches.
    const int rb = hi ? 8 : 0;
    sPD[(rb + 0) * 16 + row] = D[0];
    sPD[(rb + 1) * 16 + row] = D[1];
    sPD[(rb + 2) * 16 + row] = D[2];
    sPD[(rb + 3) * 16 + row] = D[3];
    __syncthreads();
  }

  // ---- emit new_header (before sPD is overwritten by the inverse) ----
  if (lane < 16) out[12 + lane] = sPD[(lane >> 2) * 16 + (lane & 3)];

  // ---- 4x4 inverse of new_header via adjugate (uniform, all lanes) ----
  float h[16];
#pragma unroll
  for (int r = 0; r < 4; ++r)
#pragma unroll
    for (int c = 0; c < 4; ++c)
      h[r * 4 + c] = sPD[r * 16 + c];

  float inv[16];
  inv[0]  =  h[5]*h[10]*h[15] - h[5]*h[11]*h[14] - h[9]*h[6]*h[15] + h[9]*h[7]*h[14] + h[13]*h[6]*h[11] - h[13]*h[7]*h[10];
  inv[4]  = -h[4]*h[10]*h[15] + h[4]*h[11]*h[14] + h[8]*h[6]*h[15] - h[8]*h[7]*h[14] - h[12]*h[6]*h[11] + h[12]*h[7]*h[10];
  inv[8]  =  h[4]*h[9]*h[15]  - h[4]*h[11]*h[13] - h[8]*h[5]*h[15] + h[8]*h[7]*h[13] + h[12]*h[5]*h[11] - h[12]*h[7]*h[9];
  inv[12] = -h[4]*h[9]*h[14]  + h[4]*h[10]*h[13] + h[8]*h[5]*h[14] - h[8]*h[6]*h[13] - h[12]*h[5]*h[10] + h[12]*h[6]*h[9];
  inv[1]  = -h[1]*h[10]*h[15] + h[1]*h[11]*h[14] + h[9]*h[2]*h[15] - h[9]*h[3]*h[14] - h[13]*h[2]*h[11] + h[13]*h[3]*h[10];
  inv[5]  =  h[0]*h[10]*h[15] - h[0]*h[11]*h[14] - h[8]*h[2]*h[15] + h[8]*h[3]*h[14] + h[12]*h[2]*h[11] - h[12]*h[3]*h[10];
  inv[9]  = -h[0]*h[9]*h[15]  + h[0]*h[11]*h[13] + h[8]*h[1]*h[15] - h[8]*h[3]*h[13] - h[12]*h[1]*h[11] + h[12]*h[3]*h[9];
  inv[13] =  h[0]*h[9]*h[14]  - h[0]*h[10]*h[13] - h[8]*h[1]*h[14] + h[8]*h[2]*h[13] + h[12]*h[1]*h[10] - h[12]*h[2]*h[9];
  inv[2]  =  h[1]*h[6]*h[15]  - h[1]*h[7]*h[14]  - h[5]*h[2]*h[15] + h[5]*h[3]*h[14] + h[13]*h[2]*h[7]  - h[13]*h[3]*h[6];
  inv[6]  = -h[0]*h[6]*h[15]  + h[0]*h[7]*h[14]  + h[4]*h[2]*h[15] - h[4]*h[3]*h[14] - h[12]*h[2]*h[7]  + h[12]*h[3]*h[6];
  inv[10] =  h[0]*h[5]*h[15]  - h[0]*h[7]*h[13]  - h[4]*h[1]*h[15] + h[4]*h[3]*h[13] + h[12]*h[1]*h[7]  - h[12]*h[3]*h[5];
  inv[14] = -h[0]*h[5]*h[14]  + h[0]*h[6]*h[13]  + h[4]*h[1]*h[14] - h[4]*h[2]*h[13] - h[12]*h[1]*h[6]  + h[12]*h[2]*h[5];
  inv[3]  = -h[1]*h[6]*h[11]  + h[1]*h[7]*h[10]  + h[5]*h[2]*h[11] - h[5]*h[3]*h[10] - h[9]*h[2]*h[7]   + h[9]*h[3]*h[6];
  inv[7]  =  h[0]*h[6]*h[11]  - h[0]*h[7]*h[10]  - h[4]*h[2]*h[11] + h[4]*h[3]*h[10] + h[8]*h[2]*h[7]   - h[8]*h[3]*h[6];
  inv[11] = -h[0]*h[5]*h[11]  + h[0]*h[7]*h[9]   + h[4]*h[1]*h[11] - h[4]*h[3]*h[9]  - h[8]*h[1]*h[7]   + h[8]*h[3]*h[5];
  inv[15] =  h[0]*h[5]*h[10]  - h[0]*h[6]*h[9]   - h[4]*h[1]*h[10] + h[4]*h[2]*h[9]  + h[8]*h[1]*h[6]   - h[8]*h[2]*h[5];

  float det  = h[0]*inv[0] + h[1]*inv[4] + h[2]*inv[8] + h[3]*inv[12];
  float rdet = 1.0f / det;
#pragma unroll
  for (int e = 0; e < 16; ++e) inv[e] *= rdet;

  __syncthreads();
  // lane 0 overwrites the 4x4 corner of sPD with inv (constant indices,
  // padding elsewhere still zero) so the final A-fragment load is
  // unconditional and no dynamic private-array indexing is needed.
  if (lane == 0) {
#pragma unroll
    for (int r = 0; r < 4; ++r)
#pragma unroll
      for (int c = 0; c < 4; ++c)
        sPD[r * 16 + c] = inv[r * 4 + c];
  }
  __syncthreads();

  // ---- affine_full = inv(new_header) @ vol_affine (12th WMMA) ----
  v2f A2, B2;
  A2.x = sPD[row * 16 + k0];
  A2.y = sPD[row * 16 + k1];
  B2.x = sVA[k0 * 16 + row];
  B2.y = sVA[k1 * 16 + row];
  v8f C2 = {};
  v8f D2 = __builtin_amdgcn_wmma_f32_16x16x4_f32(false, A2, false, B2,
                                                 (short)0, C2, false, false);

  // outputs: affine[1,3,4] = rows 0..2 of affine_full
  if (lane < 4) {
    out[0 * 4 + lane] = D2[0];
    out[1 * 4 + lane] = D2[1];
    out[2 * 4 + lane] = D2[2];
  }
}

extern "C" void kernel_launch(void* const* d_in, const int* in_sizes, int n_in,
                              void* d_out, int out_size, void* d_ws, size_t ws_size,
                              hipStream_t stream) {
  (void)in_sizes; (void)n_in; (void)out_size; (void)d_ws; (void)ws_size;
  const float* header      = (const float*)d_in[0];
  const float* vol_affine  = (const float*)d_in[1];
  const float* cr          = (const float*)d_in[2];
  const float* normvec     = (const float*)d_in[3];
  const float* translation = (const float*)d_in[4];
  const float* angle       = (const float*)d_in[5];
  const float* scaling     = (const float*)d_in[6];
  float* out = (float*)d_out;

  Instance3DGlobalScalingModel_kernel<<<1, 32, 0, stream>>>(
      header, vol_affine, cr, normvec, translation, angle, scaling, out);
}